// GAT_3633542332615
// MI455X (gfx1250) — compile-verified
//
#include <hip/hip_runtime.h>
#include <hip/hip_bf16.h>
#include <math.h>

// ---------------------------------------------------------------------------
// 2-layer GAT for MI455X (gfx1250, wave32).
//  - GEMM1 (x @ W1) uses V_WMMA_F32_16X16X4_F32 (full f32 precision; GEMM is
//    not the bottleneck, the edge scatter is, so no need to drop precision).
//  - Edge softmax: 3-pass (atomic max / atomic exp-sum / atomic aggregate),
//    recomputing the cheap leaky-relu term each pass to save workspace.
// ---------------------------------------------------------------------------

typedef float v2f __attribute__((ext_vector_type(2)));
typedef float v8f __attribute__((ext_vector_type(8)));

#define IN_DIM    128
#define D1        256     // HEADS*HID
#define HEADS     8
#define HID       32
#define NEG_SLOPE 0.2f
#define NEG_BIG   -3.0e38f

// -------------------------------------------------------------------- utils
__global__ void gat_fill(float* __restrict__ p, int n, float v) {
    int i = blockIdx.x * blockDim.x + threadIdx.x;
    if (i < n) p[i] = v;
}

__device__ __forceinline__ void edge_sd(const int* __restrict__ ei, int E,
                                        int e, int& s, int& d) {
    if (e < E) { s = ei[e]; d = ei[E + e]; }   // row0=src, row1=dst
    else       { s = d = e - E; }              // appended self-loop
}

// ------------------------------------------------------------- K1: WMMA GEMM
// h1[N,256] = x[N,128] @ W1[128,256], f32 WMMA 16x16x4.
// Block = 256 threads (8 waves). Block computes rows [row0,row0+16) x all 256
// cols; wave w owns cols [w*32, w*32+32) as two 16x16 accumulators.
__global__ __launch_bounds__(256) void gat_gemm1_wmma(
    const float* __restrict__ x, const float* __restrict__ W,
    float* __restrict__ h1, int N) {
    __shared__ float ws[32][D1];     // K-chunk x N slab of W1 (32 KB)
    __shared__ float xs[16][33];     // M x K-chunk tile of x (padded)

    const int tid  = threadIdx.x;
    const int lane = tid & 31;
    const int wave = tid >> 5;
    const int row0 = blockIdx.x * 16;
    const int c0   = wave * 32;

    // A-frag addressing (ISA 7.12.2, 32-bit A 16x4): lanes 0-15 -> K pair
    // {0,1}, lanes 16-31 -> K pair {2,3}; M = lane&15.
    const int koff    = (lane >> 4) * 2;
    const int halfrow = lane & 15;

    v8f acc0 = {};
    v8f acc1 = {};

    for (int kc = 0; kc < IN_DIM; kc += 32) {
        // stage W1[kc:kc+32, 0:256] -> LDS (coalesced, 32 rows / 256 threads)
#pragma unroll
        for (int i = 0; i < 32; ++i)
            ws[i][tid] = W[(kc + i) * D1 + tid];
        // stage x[row0:row0+16, kc:kc+32] -> LDS
#pragma unroll
        for (int j = 0; j < 2; ++j) {
            int idx = tid + j * 256;
            int r = idx >> 5, kk = idx & 31;
            int row = row0 + r; if (row >= N) row = N - 1;   // clamp (tail)
            xs[r][kk] = x[row * IN_DIM + kc + kk];
        }
        __syncthreads();

#pragma unroll
        for (int kk = 0; kk < 8; ++kk) {
            const int k4 = kk * 4;
            v2f a;  a.x  = xs[halfrow][k4 + koff];
                    a.y  = xs[halfrow][k4 + koff + 1];
            v2f b0; b0.x = ws[k4 + koff    ][c0 + halfrow];
                    b0.y = ws[k4 + koff + 1][c0 + halfrow];
            v2f b1; b1.x = ws[k4 + koff    ][c0 + 16 + halfrow];
                    b1.y = ws[k4 + koff + 1][c0 + 16 + halfrow];
            acc0 = __builtin_amdgcn_wmma_f32_16x16x4_f32(
                       false, a, false, b0, (short)0, acc0, false, false);
            acc1 = __builtin_amdgcn_wmma_f32_16x16x4_f32(
                       false, a, false, b1, (short)0, acc1, false, false);
        }
        __syncthreads();
    }

    // C/D layout: VGPR r -> M=r (lanes 0-15) / M=r+8 (lanes 16-31), N=lane%16
    const int mbase = (lane >> 4) * 8;
#pragma unroll
    for (int r = 0; r < 8; ++r) {
        int row = row0 + mbase + r;
        if (row < N) {
            h1[row * D1 + c0 +      halfrow] = acc0[r];
            h1[row * D1 + c0 + 16 + halfrow] = acc1[r];
        }
    }
}

// ------------------------------------------------- K2: per-node attention dots
// wave32 == one head of 32 channels: shuffle-reduce the dot products.
__global__ __launch_bounds__(256) void gat_attn1(
    const float* __restrict__ h1, const float* __restrict__ att_src,
    const float* __restrict__ att_dst, float* __restrict__ as1,
    float* __restrict__ ad1) {
    const int n = blockIdx.x, tid = threadIdx.x;
    const int head = tid >> 5, lane = tid & 31;
    float v = h1[n * D1 + tid];
    float s = v * att_src[head * HID + lane];
    float d = v * att_dst[head * HID + lane];
#pragma unroll
    for (int off = 16; off > 0; off >>= 1) {
        s += __shfl_xor(s, off, 32);
        d += __shfl_xor(d, off, 32);
    }
    if (lane == 0) { as1[n * HEADS + head] = s; ad1[n * HEADS + head] = d; }
}

// ----------------------------------------------------- K3/K4: edge max / sum
__global__ void gat_edge_max1(const int* __restrict__ ei, int E, int ET,
                              const float* __restrict__ as1,
                              const float* __restrict__ ad1,
                              float* __restrict__ m1) {
    int g = blockIdx.x * blockDim.x + threadIdx.x;
    if (g >= ET * HEADS) return;
    int e = g >> 3, h = g & 7, s, d;
    edge_sd(ei, E, e, s, d);
    float t = as1[s * HEADS + h] + ad1[d * HEADS + h];
    t = t > 0.f ? t : NEG_SLOPE * t;
    atomicMax(&m1[d * HEADS + h], t);
}

__global__ void gat_edge_sum1(const int* __restrict__ ei, int E, int ET,
                              const float* __restrict__ as1,
                              const float* __restrict__ ad1,
                              const float* __restrict__ m1,
                              float* __restrict__ den1) {
    int g = blockIdx.x * blockDim.x + threadIdx.x;
    if (g >= ET * HEADS) return;
    int e = g >> 3, h = g & 7, s, d;
    edge_sd(ei, E, e, s, d);
    float t = as1[s * HEADS + h] + ad1[d * HEADS + h];
    t = t > 0.f ? t : NEG_SLOPE * t;
    atomicAdd(&den1[d * HEADS + h], expf(t - m1[d * HEADS + h]));
}

// ------------------------------------------------------ K5: edge aggregation
// One block per edge; lane c gathers h1[src,c], scatter-adds alpha*h into dst.
__global__ __launch_bounds__(256) void gat_edge_agg1(
    const int* __restrict__ ei, int E,
    const float* __restrict__ as1, const float* __restrict__ ad1,
    const float* __restrict__ m1, const float* __restrict__ den1,
    const float* __restrict__ h1, float* __restrict__ agg1) {
    int e = blockIdx.x, c = threadIdx.x, h = c >> 5, s, d;
    edge_sd(ei, E, e, s, d);
    float t = as1[s * HEADS + h] + ad1[d * HEADS + h];
    t = t > 0.f ? t : NEG_SLOPE * t;
    float alpha = expf(t - m1[d * HEADS + h]) / den1[d * HEADS + h];
    atomicAdd(&agg1[d * D1 + c], alpha * h1[s * D1 + c]);
}

// ----------------------- K6: bias + ELU + GEMM2 (256->2) + layer-2 attn dots
__global__ __launch_bounds__(256) void gat_l1fin_gemm2(
    const float* __restrict__ agg1, const float* __restrict__ b1,
    const float* __restrict__ W2, const float* __restrict__ att_src2,
    const float* __restrict__ att_dst2, float* __restrict__ z,
    float* __restrict__ as2, float* __restrict__ ad2) {
    __shared__ float s0[256], s1[256];
    const int n = blockIdx.x, t = threadIdx.x;
    float v = agg1[n * D1 + t] + b1[t];
    v = v > 0.f ? v : expm1f(v);                 // ELU(alpha=1)
    s0[t] = v * W2[t * 2 + 0];
    s1[t] = v * W2[t * 2 + 1];
    __syncthreads();
    for (int off = 128; off > 0; off >>= 1) {
        if (t < off) { s0[t] += s0[t + off]; s1[t] += s1[t + off]; }
        __syncthreads();
    }
    if (t == 0) {
        float z0 = s0[0], z1 = s1[0];
        z[n * 2 + 0] = z0; z[n * 2 + 1] = z1;
        as2[n] = z0 * att_src2[0] + z1 * att_src2[1];
        ad2[n] = z0 * att_dst2[0] + z1 * att_dst2[1];
    }
}

// ----------------------------------------------- K7-K9: layer-2 edge softmax
__global__ void gat_edge_max2(const int* __restrict__ ei, int E, int ET,
                              const float* __restrict__ as2,
                              const float* __restrict__ ad2,
                              float* __restrict__ m2) {
    int e = blockIdx.x * blockDim.x + threadIdx.x;
    if (e >= ET) return;
    int s, d; edge_sd(ei, E, e, s, d);
    float t = as2[s] + ad2[d];
    t = t > 0.f ? t : NEG_SLOPE * t;
    atomicMax(&m2[d], t);
}

__global__ void gat_edge_sum2(const int* __restrict__ ei, int E, int ET,
                              const float* __restrict__ as2,
                              const float* __restrict__ ad2,
                              const float* __restrict__ m2,
                              float* __restrict__ den2) {
    int e = blockIdx.x * blockDim.x + threadIdx.x;
    if (e >= ET) return;
    int s, d; edge_sd(ei, E, e, s, d);
    float t = as2[s] + ad2[d];
    t = t > 0.f ? t : NEG_SLOPE * t;
    atomicAdd(&den2[d], expf(t - m2[d]));
}

__global__ void gat_edge_agg2(const int* __restrict__ ei, int E, int ET,
                              const float* __restrict__ as2,
                              const float* __restrict__ ad2,
                              const float* __restrict__ m2,
                              const float* __restrict__ den2,
                              const float* __restrict__ z,
                              float* __restrict__ agg2) {
    int e = blockIdx.x * blockDim.x + threadIdx.x;
    if (e >= ET) return;
    int s, d; edge_sd(ei, E, e, s, d);
    float t = as2[s] + ad2[d];
    t = t > 0.f ? t : NEG_SLOPE * t;
    float alpha = expf(t - m2[d]) / den2[d];
    atomicAdd(&agg2[d * 2 + 0], alpha * z[s * 2 + 0]);
    atomicAdd(&agg2[d * 2 + 1], alpha * z[s * 2 + 1]);
}

// ----------------------------------------------------- K10: bias+log_softmax
__global__ void gat_logsoftmax(const float* __restrict__ agg2,
                               const float* __restrict__ b2,
                               float* __restrict__ out, int N) {
    int n = blockIdx.x * blockDim.x + threadIdx.x;
    if (n >= N) return;
    float o0 = agg2[n * 2 + 0] + b2[0];
    float o1 = agg2[n * 2 + 1] + b2[1];
    float mx = fmaxf(o0, o1);
    float lse = mx + logf(expf(o0 - mx) + expf(o1 - mx));
    out[n * 2 + 0] = o0 - lse;
    out[n * 2 + 1] = o1 - lse;
}

// ---------------------------------------------------------------------------
extern "C" void kernel_launch(void* const* d_in, const int* in_sizes, int n_in,
                              void* d_out, int out_size, void* d_ws, size_t ws_size,
                              hipStream_t stream) {
    const float* x        = (const float*)d_in[0];
    const int*   ei       = (const int*)  d_in[1];   // [2,E] flat: src row, dst row
    const float* W1       = (const float*)d_in[2];
    const float* att_src1 = (const float*)d_in[3];
    const float* att_dst1 = (const float*)d_in[4];
    const float* b1       = (const float*)d_in[5];
    const float* W2       = (const float*)d_in[6];
    const float* att_src2 = (const float*)d_in[7];
    const float* att_dst2 = (const float*)d_in[8];
    const float* b2       = (const float*)d_in[9];

    const int N  = in_sizes[0] / IN_DIM;
    const int E  = in_sizes[1] / 2;
    const int ET = E + N;                     // edges + self-loops

    // ---- workspace layout (floats)
    // NOTE: m1..m2 form one contiguous -inf-filled group (N*9 floats) and
    // agg1..den2 form one contiguous zero-filled group (N*267 floats).
    float* p    = (float*)d_ws;
    float* h1   = p;  p += (size_t)N * D1;
    float* as1  = p;  p += (size_t)N * HEADS;
    float* ad1  = p;  p += (size_t)N * HEADS;
    float* z    = p;  p += (size_t)N * 2;
    float* as2  = p;  p += (size_t)N;
    float* ad2  = p;  p += (size_t)N;
    float* m1   = p;  p += (size_t)N * HEADS;     // -inf group begin
    float* m2   = p;  p += (size_t)N;             // -inf group end
    float* agg1 = p;  p += (size_t)N * D1;        // zero group begin
    float* den1 = p;  p += (size_t)N * HEADS;
    float* agg2 = p;  p += (size_t)N * 2;
    float* den2 = p;  p += (size_t)N;             // zero group end

    const int TB = 256;
    const int nInf  = N * (HEADS + 1);
    const int nZero = N * (D1 + HEADS + 3);
    gat_fill<<<(nInf  + TB - 1) / TB, TB, 0, stream>>>(m1,   nInf,  NEG_BIG);
    gat_fill<<<(nZero + TB - 1) / TB, TB, 0, stream>>>(agg1, nZero, 0.0f);

    // layer 1
    gat_gemm1_wmma<<<(N + 15) / 16, TB, 0, stream>>>(x, W1, h1, N);
    gat_attn1<<<N, TB, 0, stream>>>(h1, att_src1, att_dst1, as1, ad1);
    const int EH = ET * HEADS;
    gat_edge_max1<<<(EH + TB - 1) / TB, TB, 0, stream>>>(ei, E, ET, as1, ad1, m1);
    gat_edge_sum1<<<(EH + TB - 1) / TB, TB, 0, stream>>>(ei, E, ET, as1, ad1, m1, den1);
    gat_edge_agg1<<<ET, TB, 0, stream>>>(ei, E, as1, ad1, m1, den1, h1, agg1);

    // finalize layer 1 + layer-2 projection
    gat_l1fin_gemm2<<<N, TB, 0, stream>>>(agg1, b1, W2, att_src2, att_dst2,
                                          z, as2, ad2);
    // layer 2
    gat_edge_max2<<<(ET + TB - 1) / TB, TB, 0, stream>>>(ei, E, ET, as2, ad2, m2);
    gat_edge_sum2<<<(ET + TB - 1) / TB, TB, 0, stream>>>(ei, E, ET, as2, ad2, m2, den2);
    gat_edge_agg2<<<(ET + TB - 1) / TB, TB, 0, stream>>>(ei, E, ET, as2, ad2, m2,
                                                         den2, z, agg2);
    gat_logsoftmax<<<(N + TB - 1) / TB, TB, 0, stream>>>(agg2, b2,
                                                         (float*)d_out, N);
}